// GaborLayer_1898375545669
// MI455X (gfx1250) — compile-verified
//
#include <hip/hip_runtime.h>
#include <cstdint>

// ---------------------------------------------------------------------------
// Gabor layer forward for MI455X (gfx1250, wave32).
//   out[3,512,512] = clip( sum_g amp * exp(-xr^2/2s^2 - yr^2/2g^2) *
//                                cos(freq*xr + phase), -1, 1 )
// Trans-pipe-bound (~67M exp + ~134M sin/cos); FP32 throughout.
//   k1: per-gabor derived table (16 f32/gabor) -> d_ws (16 KB)
//       - revolution domain for hw v_cos/v_sin (cos(2*pi*x))
//       - log2e folded into gaussian coefficients for hw v_exp (2^x)
//       - amp*cos(2*pi*psi), -amp*sin(2*pi*psi) folded for angle-addition
//   k2: TDM tensor_load_to_lds stages table into LDS once per block
//       (s_wait_tensorcnt + barrier), then 2 pixels/thread integrate over
//       256 gabors with ds_load_b128 broadcasts; exact range reduction via
//       v_fract_f32 keeps hw sin/cos in their valid domain.
// ---------------------------------------------------------------------------

typedef float        f32x4 __attribute__((ext_vector_type(4)));
typedef unsigned int u32x4 __attribute__((ext_vector_type(4)));
typedef int          i32x4v __attribute__((ext_vector_type(4)));
typedef int          i32x8v __attribute__((ext_vector_type(8)));

#define NGABOR      256
#define IMG_H       512
#define IMG_W       512
#define HW          (IMG_H * IMG_W)
#define PPG         16                      // padded floats per gabor
#define TBL_FLOATS  (NGABOR * PPG)          // 4096
#define TBL_BYTES   (TBL_FLOATS * 4)        // 16384
#define PX_PER_THREAD 2
#define BLOCK_PX    (256 * PX_PER_THREAD)   // 512 pixels per block

__device__ __forceinline__ float clampf(float x, float lo, float hi) {
    return fminf(fmaxf(x, lo), hi);
}

// ---------------------------------------------------------------------------
// Kernel 1: derived parameter table.
// layout per gabor g (16 floats at tbl[g*16]):
//   [0]=u  [1]=v  [2]=cos(2pi*theta_rev) [3]=sin(2pi*theta_rev)
//   [4]=-log2e/(2*sigma^2)  [5]=-log2e/(2*gamma^2)  [6]=exp(-rel_freq) [7]=pad
//   [8..10]  =  amp_c *  cos(2pi*psi_c)
//   [12..14] = -amp_c *  sin(2pi*psi_c)
// ---------------------------------------------------------------------------
__global__ __launch_bounds__(NGABOR)
void gabor_precompute(const float* __restrict__ u,  const float* __restrict__ v,
                      const float* __restrict__ th, const float* __restrict__ rs,
                      const float* __restrict__ rf, const float* __restrict__ gm,
                      const float* __restrict__ psi, const float* __restrict__ amp,
                      float* __restrict__ tbl) {
    const int g = threadIdx.x;

    const float uc  = clampf(u[g],  -1.0f, 1.0f);
    const float vc  = clampf(v[g],  -1.0f, 1.0f);
    const float thc = clampf(th[g], -2.0f, 2.0f);      // revolutions
    const float sig = clampf(rs[g], 0.001f, 1.0f);
    const float gam = clampf(gm[g], 0.0001f, 1.0f);
    const float rfc = clampf(rf[g], -5.0f, 5.0f);

    const float cr = __builtin_amdgcn_cosf(thc);       // cos(2*pi*thc)
    const float sr = __builtin_amdgcn_sinf(thc);

    const float HALF_LOG2E = 0.721347520444481703f;    // log2(e)/2
    const float An = -HALF_LOG2E * __builtin_amdgcn_rcpf(sig * sig);
    const float Bn = -HALF_LOG2E * __builtin_amdgcn_rcpf(gam * gam);

    const float LOG2E = 1.44269504088896341f;
    const float frev  = __builtin_amdgcn_exp2f(-rfc * LOG2E); // freq/(2*pi)

    f32x4 q0, q1, q2, q3;
    q0.x = uc; q0.y = vc; q0.z = cr; q0.w = sr;
    q1.x = An; q1.y = Bn; q1.z = frev; q1.w = 0.0f;
#pragma unroll
    for (int c = 0; c < 3; ++c) {
        const float ph = clampf(psi[g * 3 + c], -1.0f, 1.0f);  // revolutions
        const float am = clampf(amp[g * 3 + c], 0.0f, 1.0f);
        const float cp =  am * __builtin_amdgcn_cosf(ph);
        const float sp = -am * __builtin_amdgcn_sinf(ph);
        if (c == 0) { q2.x = cp; q3.x = sp; }
        if (c == 1) { q2.y = cp; q3.y = sp; }
        if (c == 2) { q2.z = cp; q3.z = sp; }
    }
    q2.w = 0.0f; q3.w = 0.0f;

    f32x4* o = (f32x4*)(tbl + g * PPG);
    o[0] = q0; o[1] = q1; o[2] = q2; o[3] = q3;
}

// ---------------------------------------------------------------------------
// Kernel 2: main integration. 512 blocks x 256 threads, 2 pixels/thread.
// ---------------------------------------------------------------------------
__global__ __launch_bounds__(256)
void gabor_main(const float* __restrict__ gx, const float* __restrict__ gy,
                const float* __restrict__ tbl, float* __restrict__ out) {
    extern __shared__ float sp[];           // TBL_BYTES of dynamic LDS, offset 0

    // ---- Stage parameter table into LDS via the Tensor Data Mover ----------
#if __has_builtin(__builtin_amdgcn_tensor_load_to_lds)
    if (threadIdx.x < 32) {                 // wave 0 only issues the DMA
        const unsigned long long ga = (unsigned long long)(uintptr_t)tbl;
        u32x4 g0;
        g0.x = 1u;                                          // count=1, user mode
        g0.y = 0u;                                          // lds_addr = 0
        g0.z = (unsigned)(ga & 0xFFFFFFFFull);              // global_addr[31:0]
        g0.w = (unsigned)((ga >> 32) & 0x1FFFFFFull) | (2u << 30); // [56:32]+type=2

        i32x8v g1;
        g1[0] = (int)(2u << 16);                            // data_size = 4B
        g1[1] = (int)((TBL_FLOATS & 0xFFFF) << 16);         // tensor_dim0 lo16
        g1[2] = (int)(((TBL_FLOATS >> 16) & 0xFFFF) | (1u << 16)); // dim0 hi / dim1=1
        g1[3] = (int)((TBL_FLOATS & 0xFFFF) << 16);         // tile_dim0 = 4096
        g1[4] = 1;                                          // tile_dim1 = 1
        g1[5] = TBL_FLOATS;                                 // tensor_dim0_stride
        g1[6] = 0;
        g1[7] = 0;

        i32x4v z4 = {0, 0, 0, 0};
#if __clang_major__ >= 23
        i32x8v z8 = {0, 0, 0, 0, 0, 0, 0, 0};
        __builtin_amdgcn_tensor_load_to_lds(g0, g1, z4, z4, z8, 0);
#else
        __builtin_amdgcn_tensor_load_to_lds(g0, g1, z4, z4, 0);
#endif
#if __has_builtin(__builtin_amdgcn_s_wait_tensorcnt)
        __builtin_amdgcn_s_wait_tensorcnt(0);
#else
        asm volatile("s_wait_tensorcnt 0" ::: "memory");
#endif
    }
#else
    // Fallback: cooperative copy (still lands in LDS).
    for (int i = threadIdx.x; i < TBL_FLOATS; i += 256) sp[i] = tbl[i];
#endif
    __syncthreads();

    // ---- Integrate PX_PER_THREAD pixels per thread over all gabors ---------
    const int t    = threadIdx.x;
    const int base = blockIdx.x * BLOCK_PX + t;

    float X[PX_PER_THREAD], Y[PX_PER_THREAD];
    float a0[PX_PER_THREAD], a1[PX_PER_THREAD], a2[PX_PER_THREAD];

#pragma unroll
    for (int i = 0; i < PX_PER_THREAD; ++i) {
        const int p = base + i * 256;
        X[i] = gx[p];
        Y[i] = gy[p];
        a0[i] = 0.0f; a1[i] = 0.0f; a2[i] = 0.0f;
    }

    for (int g = 0; g < NGABOR; ++g) {
        const f32x4* q = (const f32x4*)(sp + (g << 4));   // uniform addr -> LDS bcast
        const f32x4 pa = q[0];   // u, v, cr, sr
        const f32x4 pb = q[1];   // An, Bn, frev, -
        const f32x4 pc = q[2];   // amp*cos(phase)  per channel
        const f32x4 pd = q[3];   // -amp*sin(phase) per channel

#pragma unroll
        for (int i = 0; i < PX_PER_THREAD; ++i) {
            const float dx = X[i] - pa.x;
            const float dy = Y[i] - pa.y;
            const float xr = fmaf(dx, pa.z, dy * pa.w);    //  dx*cr + dy*sr
            const float yr = fmaf(dy, pa.z, -(dx * pa.w)); // -dx*sr + dy*cr
            const float e  = fmaf(xr * xr, pb.x, (yr * yr) * pb.y); // negative
            const float ga = __builtin_amdgcn_exp2f(e);    // gaussian
            // base in revolutions; exact range reduction via periodicity
            const float br = __builtin_amdgcn_fractf(xr * pb.z);   // [0,1)
            const float cb = __builtin_amdgcn_cosf(br);
            const float sb = __builtin_amdgcn_sinf(br);
            const float gc = ga * cb;
            const float gs = ga * sb;
            // cos(base+phase) folded: amp*ga*cos(.) = pc*gc + pd*gs
            a0[i] = fmaf(pc.x, gc, a0[i]); a0[i] = fmaf(pd.x, gs, a0[i]);
            a1[i] = fmaf(pc.y, gc, a1[i]); a1[i] = fmaf(pd.y, gs, a1[i]);
            a2[i] = fmaf(pc.z, gc, a2[i]); a2[i] = fmaf(pd.z, gs, a2[i]);
        }
    }

#pragma unroll
    for (int i = 0; i < PX_PER_THREAD; ++i) {
        const int p = base + i * 256;
        out[p]          = clampf(a0[i], -1.0f, 1.0f);
        out[HW + p]     = clampf(a1[i], -1.0f, 1.0f);
        out[2 * HW + p] = clampf(a2[i], -1.0f, 1.0f);
    }
}

// ---------------------------------------------------------------------------
extern "C" void kernel_launch(void* const* d_in, const int* in_sizes, int n_in,
                              void* d_out, int out_size, void* d_ws, size_t ws_size,
                              hipStream_t stream) {
    (void)in_sizes; (void)n_in; (void)out_size; (void)ws_size;
    const float* gx  = (const float*)d_in[0];
    const float* gy  = (const float*)d_in[1];
    const float* u   = (const float*)d_in[2];
    const float* v   = (const float*)d_in[3];
    const float* th  = (const float*)d_in[4];
    const float* rs  = (const float*)d_in[5];
    const float* rf  = (const float*)d_in[6];
    const float* gm  = (const float*)d_in[7];
    const float* psi = (const float*)d_in[8];
    const float* amp = (const float*)d_in[9];

    float* tbl = (float*)d_ws;
    float* out = (float*)d_out;

    gabor_precompute<<<1, NGABOR, 0, stream>>>(u, v, th, rs, rf, gm, psi, amp, tbl);
    gabor_main<<<HW / BLOCK_PX, 256, TBL_BYTES, stream>>>(gx, gy, tbl, out);
}